// OSARLayer_34428457845503
// MI455X (gfx1250) — compile-verified
//
#include <hip/hip_runtime.h>

// OSAR layer for MI455X (gfx1250, wave32).
//
// Reference decomposition:
//   out[b, 0:2016]      = memory[b, 32:2048]              (copy)
//   out[b, 2016:2048]   = conv1d(memory[b,2048:2176], W)  (GEMM, WMMA f32)
//   out[b, 2048:3968]   = memory[b, 2176:4096]            (copy)
//   out[b, 3968:4096]   = inputs[b, :]                    (copy)
//
// Conv-as-GEMM: M = 32 batches * 32 out-positions = 1024, K = 4*1024 = 4096
// (A rows are contiguous in memory!), N = 1024. W (RATE,F,F) is already a
// contiguous row-major K x N matrix. fp32 WMMA 16x16x4 keeps exact precision.

typedef __attribute__((ext_vector_type(2))) float v2f;
typedef __attribute__((ext_vector_type(8))) float v8f;

#define NB      32
#define NF      1024
#define NUNITS  4096        // conv_units + short_units
#define TC      32          // compression length per batch
#define KDIM    4096        // RATE * F
#define ROW4    256         // float4 per feature row
#define CROWS   4064        // copied unit-rows per batch (4096 - 32 compression rows)

// ---------------------------------------------------------------------------
// Bulk data movement: everything except the 32 compression rows per batch.
// float4 loads/stores -> global_load_b128 / global_store_b128, HBM-bound.
// ---------------------------------------------------------------------------
__global__ __launch_bounds__(256) void osar_copy_kernel(
    const float4* __restrict__ mem4,
    const float4* __restrict__ in4,
    float4* __restrict__ out4)
{
  long idx = (long)blockIdx.x * blockDim.x + threadIdx.x;
  const long total = (long)NB * CROWS * ROW4;
  if (idx >= total) return;

  const int  f4  = (int)(idx & (ROW4 - 1));
  const long row = idx >> 8;                 // ROW4 == 256
  const int  b   = (int)(row / CROWS);
  const int  ur  = (int)(row % CROWS);

  const long MEMB = (long)NUNITS * ROW4;     // float4 per batch of memory/out
  const long obase = (long)b * MEMB;

  if (ur < 2016) {
    // new_conv head: conv_mem[:, 32:2048]
    out4[obase + (long)ur * ROW4 + f4] =
        mem4[obase + (long)(32 + ur) * ROW4 + f4];
  } else if (ur < 3936) {
    // new_short head: short_mem[:, 128:2048]  (dst u = 2048+j, src u = dst+128)
    const int du = 2048 + (ur - 2016);
    out4[obase + (long)du * ROW4 + f4] =
        mem4[obase + (long)(du + 128) * ROW4 + f4];
  } else {
    // new_short tail: inputs
    const int j = ur - 3936;                 // 0..127
    out4[obase + (long)(3968 + j) * ROW4 + f4] =
        in4[(long)b * (128L * ROW4) + (long)j * ROW4 + f4];
  }
}

// ---------------------------------------------------------------------------
// Compression GEMM via V_WMMA_F32_16X16X4_F32.
// One wave (32 lanes) computes one 16x16 tile of C; 4 waves/block span N.
// A-fragment (16x4 f32): lane L -> row m = L&15, K-pair kOff = 2*(L>>4).
// B-fragment (4x16 f32): lane L -> col n = L&15, same kOff.
// C/D (16x16 f32, 8 VGPRs): element (v + 8*(L>>4), L&15).
// ---------------------------------------------------------------------------
__global__ __launch_bounds__(128) void osar_conv_wmma_kernel(
    const float* __restrict__ mem,
    const float* __restrict__ w,
    const float* __restrict__ bias,
    float* __restrict__ out)
{
  const int lane  = threadIdx.x & 31;
  const int wave  = threadIdx.x >> 5;
  const int tileN = blockIdx.x * 4 + wave;   // 0..63
  const int tileM = blockIdx.y;              // 0..63

  const int mr    = lane & 15;               // row within tile (A/C half-layout)
  const int kHalf = (lane >> 4) * 2;         // 0 or 2

  // Global M row for the A fragment this lane feeds.
  const int m  = tileM * 16 + mr;            // 0..1023
  const int b  = m >> 5;                     // batch
  const int i  = m & 31;                     // output position in batch
  // A row = memory[b, 2048 + 4*i, 0 .. 4095]  (4 consecutive unit-rows)
  const float* __restrict__ arow =
      mem + (size_t)b * ((size_t)NUNITS * NF) + (size_t)(2048 + 4 * i) * NF;

  const int n = tileN * 16 + (lane & 15);    // column for B/C fragments
  const float* __restrict__ bcol = w + n;    // W[k*NF + n]

  v8f acc = {};
  #pragma unroll 4
  for (int k = 0; k < KDIM; k += 4) {
    // Keep the A stream ahead in cache (emits global_prefetch_b8 on gfx1250).
    __builtin_prefetch(arow + k + 128, 0, 1);

    v2f a;
    a.x = arow[k + kHalf];
    a.y = arow[k + kHalf + 1];

    v2f bb;
    bb.x = bcol[(size_t)(k + kHalf) * NF];
    bb.y = bcol[(size_t)(k + kHalf + 1) * NF];

    // D = A(16x4) * B(4x16) + C, fp32 throughout.
    acc = __builtin_amdgcn_wmma_f32_16x16x4_f32(
        /*neg_a=*/false, a, /*neg_b=*/false, bb,
        /*c_mod=*/(short)0, acc, /*reuse_a=*/false, /*reuse_b=*/false);
  }

  const float bv = bias[n];
  const int rowBase = (lane >> 4) * 8;
  #pragma unroll
  for (int v = 0; v < 8; ++v) {
    const int gm = tileM * 16 + rowBase + v; // global M row of acc[v]
    const int gb = gm >> 5;
    const int gi = gm & 31;
    out[(size_t)gb * ((size_t)NUNITS * NF) + (size_t)(2016 + gi) * NF + n] =
        acc[v] + bv;
  }
}

extern "C" void kernel_launch(void* const* d_in, const int* in_sizes, int n_in,
                              void* d_out, int out_size, void* d_ws, size_t ws_size,
                              hipStream_t stream) {
  const float* inputs = (const float*)d_in[0];   // (32, 128, 1024)
  const float* memory = (const float*)d_in[1];   // (32, 4096, 1024)
  const float* conv_w = (const float*)d_in[2];   // (4, 1024, 1024)
  const float* conv_b = (const float*)d_in[3];   // (1024,)
  // d_in[4..6] = conv_units / short_units / rate scalars (compile-time here)
  float* out = (float*)d_out;                    // (32, 4096, 1024)

  // Bulk copies (HBM-bound, ~1 GB traffic -> ~44 us at 23.3 TB/s)
  const long n4 = (long)NB * CROWS * ROW4;
  const int blocks = (int)((n4 + 255) / 256);
  osar_copy_kernel<<<blocks, 256, 0, stream>>>(
      (const float4*)memory, (const float4*)inputs, (float4*)out);

  // Compression GEMM: 64 N-tiles x 64 M-tiles of 16x16, 4 waves per block.
  dim3 grid(16, 64);
  osar_conv_wmma_kernel<<<grid, 128, 0, stream>>>(memory, conv_w, conv_b, out);
}